// MPConv_layer_relu_82188494176500
// MI455X (gfx1250) — compile-verified
//
#include <hip/hip_runtime.h>

// Message passing with mean aggregation:
//   out[i] = mean over edges (j -> i) of x[j]; empty neighborhoods -> 0
// N = 100000 nodes, E = 800000 edges, F = 64 features (float32).
//
// Strategy: on-device counting sort of edges by target (CSR build), then a
// per-node wave32 gather with register accumulation -> zero float atomics,
// single streaming write of out. Fallback to the float-atomic scatter path
// if d_ws is too small for the CSR scratch.

#define F_DIM 64
#define SCAN_CHUNK 1024

// ===========================================================================
// CSR-build pipeline
// ===========================================================================

// --- zero counts + cursors -------------------------------------------------
__global__ void mp_zero_kernel(int* __restrict__ cnt,
                               int* __restrict__ cursor,
                               int n_nodes) {
    int i = blockIdx.x * blockDim.x + threadIdx.x;
    if (i < n_nodes) {
        cnt[i]    = 0;
        cursor[i] = 0;
    }
}

// --- histogram of edge targets --------------------------------------------
__global__ void mp_count_kernel(const int* __restrict__ tgt,
                                int* __restrict__ cnt,
                                int n_edges) {
    int e = blockIdx.x * blockDim.x + threadIdx.x;
    if (e < n_edges) {
        __hip_atomic_fetch_add(cnt + tgt[e], 1, __ATOMIC_RELAXED,
                               __HIP_MEMORY_SCOPE_AGENT);
    }
}

// --- scan step 1: per-chunk exclusive scan (Hillis-Steele in LDS) ----------
__global__ void mp_scan_chunk_kernel(const int* __restrict__ cnt,
                                     int* __restrict__ rowptr,
                                     int* __restrict__ chunk_sums,
                                     int n) {
    __shared__ int tmp[SCAN_CHUNK];
    const int tid = threadIdx.x;
    const int gid = blockIdx.x * SCAN_CHUNK + tid;

    int v = (gid < n) ? cnt[gid] : 0;
    tmp[tid] = v;
    __syncthreads();

    #pragma unroll
    for (int off = 1; off < SCAN_CHUNK; off <<= 1) {
        int t = (tid >= off) ? tmp[tid - off] : 0;
        __syncthreads();
        tmp[tid] += t;
        __syncthreads();
    }

    if (gid < n) rowptr[gid] = tmp[tid] - v;          // exclusive
    if (tid == SCAN_CHUNK - 1) chunk_sums[blockIdx.x] = tmp[tid];
}

// --- scan step 2: exclusive scan of chunk sums (single block, B <= 256) ----
__global__ void mp_scan_sums_kernel(int* __restrict__ sums, int n_chunks) {
    __shared__ int tmp[256];
    const int tid = threadIdx.x;
    int v = (tid < n_chunks) ? sums[tid] : 0;
    tmp[tid] = v;
    __syncthreads();

    #pragma unroll
    for (int off = 1; off < 256; off <<= 1) {
        int t = (tid >= off) ? tmp[tid - off] : 0;
        __syncthreads();
        tmp[tid] += t;
        __syncthreads();
    }

    if (tid < n_chunks) sums[tid] = tmp[tid] - v;     // exclusive, in place
}

// --- scan step 3: add chunk offsets ---------------------------------------
__global__ void mp_scan_add_kernel(int* __restrict__ rowptr,
                                   const int* __restrict__ chunk_offs,
                                   int n) {
    int gid = blockIdx.x * SCAN_CHUNK + threadIdx.x;
    if (gid < n) rowptr[gid] += chunk_offs[blockIdx.x];
}

// --- CSR fill: bucket src ids by target ------------------------------------
__global__ void mp_fill_kernel(const int* __restrict__ src,
                               const int* __restrict__ tgt,
                               const int* __restrict__ rowptr,
                               int* __restrict__ cursor,
                               int* __restrict__ csr,
                               int n_edges) {
    int e = blockIdx.x * blockDim.x + threadIdx.x;
    if (e < n_edges) {
        int d   = tgt[e];
        int pos = __hip_atomic_fetch_add(cursor + d, 1, __ATOMIC_RELAXED,
                                         __HIP_MEMORY_SCOPE_AGENT);
        csr[rowptr[d] + pos] = src[e];
    }
}

// --- per-node gather + mean: one wave32 per node ---------------------------
// Batch-load up to 32 neighbor ids coalesced; broadcast via lane permute;
// 2 neighbors/iteration: lanes 0-15 handle neighbor j, lanes 16-31 neighbor
// j+1, each lane a float4 slice (global_load_b128 gather from L2-resident x).
// Prefetch 4 neighbors ahead. Cross-half shuffle reduce, divide, one
// coalesced 256B store per node. No float atomics, no init/finalize passes.
__global__ void mp_gather_kernel(const float* __restrict__ x,
                                 const int* __restrict__ csr,
                                 const int* __restrict__ rowptr,
                                 const int* __restrict__ cnt,
                                 float* __restrict__ out,
                                 int n_nodes) {
    const int lane = threadIdx.x & 31;
    const int node = (blockIdx.x * blockDim.x + threadIdx.x) >> 5;
    if (node >= n_nodes) return;

    const int row = rowptr[node];       // broadcast loads (wave-uniform addr)
    const int deg = cnt[node];

    const int half = lane >> 4;         // pair slot: 0 or 1
    const int fofs = (lane & 15) << 2;  // float4 slice within the 64-f row

    float4 acc = make_float4(0.f, 0.f, 0.f, 0.f);

    for (int k = 0; k < deg; k += 32) {
        int m = deg - k;
        if (m > 32) m = 32;

        int nb = 0;
        if (lane < m) nb = csr[row + k + lane];       // coalesced id batch

        for (int j = 0; j < m; j += 2) {
            int jp = j + 4 + half;                    // prefetch ahead
            if (jp < m) {
                int sp = __shfl(nb, jp);
                __builtin_prefetch(x + (size_t)sp * F_DIM + fofs, 0, 3);
            }
            int jj = j + half;
            if (jj < m) {
                int s = __shfl(nb, jj);
                const float4 v = *reinterpret_cast<const float4*>(
                    x + (size_t)s * F_DIM + fofs);
                acc.x += v.x;
                acc.y += v.y;
                acc.z += v.z;
                acc.w += v.w;
            }
        }
    }

    // Combine the two half-wave partials (lane L <-> lane L^16).
    acc.x += __shfl(acc.x, lane ^ 16);
    acc.y += __shfl(acc.y, lane ^ 16);
    acc.z += __shfl(acc.z, lane ^ 16);
    acc.w += __shfl(acc.w, lane ^ 16);

    if (lane < 16) {
        float denom = (deg > 1) ? (float)deg : 1.f;
        float4 r;
        r.x = acc.x / denom;
        r.y = acc.y / denom;
        r.z = acc.z / denom;
        r.w = acc.w / denom;
        *reinterpret_cast<float4*>(out + (size_t)node * F_DIM + fofs) = r;
    }
}

// ===========================================================================
// Fallback pipeline (float atomics) if d_ws is too small for CSR scratch.
// ===========================================================================
__global__ void mp_init_kernel(float4* __restrict__ out4,
                               int* __restrict__ cnt,
                               int n4, int n_nodes) {
    int i = blockIdx.x * blockDim.x + threadIdx.x;
    if (i < n4)      out4[i] = make_float4(0.f, 0.f, 0.f, 0.f);
    if (i < n_nodes) cnt[i]  = 0;
}

__global__ void mp_edge_scatter_kernel(const float* __restrict__ x,
                                       const int* __restrict__ src,
                                       const int* __restrict__ tgt,
                                       float* __restrict__ out,
                                       int* __restrict__ cnt,
                                       int n_edges) {
    const int lane = threadIdx.x & 31;
    const int wave = (blockIdx.x * blockDim.x + threadIdx.x) >> 5;
    const int base = wave << 5;
    if (base >= n_edges) return;

    int sv = 0, dv = 0;
    if (base + lane < n_edges) {
        sv = src[base + lane];
        dv = tgt[base + lane];
    }

    const int half = lane >> 4;
    const int fofs = (lane & 15) << 2;

    #pragma unroll 4
    for (int j = 0; j < 32; j += 2) {
        int e = base + j + half;
        if (e < n_edges) {
            int s = __shfl(sv, j + half);
            int d = __shfl(dv, j + half);
            const float4 v = *reinterpret_cast<const float4*>(
                x + (size_t)s * F_DIM + fofs);
            float* dst = out + (size_t)d * F_DIM + fofs;
            __hip_atomic_fetch_add(dst + 0, v.x, __ATOMIC_RELAXED,
                                   __HIP_MEMORY_SCOPE_AGENT);
            __hip_atomic_fetch_add(dst + 1, v.y, __ATOMIC_RELAXED,
                                   __HIP_MEMORY_SCOPE_AGENT);
            __hip_atomic_fetch_add(dst + 2, v.z, __ATOMIC_RELAXED,
                                   __HIP_MEMORY_SCOPE_AGENT);
            __hip_atomic_fetch_add(dst + 3, v.w, __ATOMIC_RELAXED,
                                   __HIP_MEMORY_SCOPE_AGENT);
            if ((lane & 15) == 0) {
                __hip_atomic_fetch_add(cnt + d, 1, __ATOMIC_RELAXED,
                                       __HIP_MEMORY_SCOPE_AGENT);
            }
        }
    }
}

__global__ void mp_finalize_kernel(float4* __restrict__ out4,
                                   const int* __restrict__ cnt,
                                   int n4) {
    int i = blockIdx.x * blockDim.x + threadIdx.x;
    if (i >= n4) return;
    int node = i >> 4;
    float c = (float)cnt[node];
    float denom = c > 1.f ? c : 1.f;
    float4 v = out4[i];
    v.x /= denom;
    v.y /= denom;
    v.z /= denom;
    v.w /= denom;
    out4[i] = v;
}

// ===========================================================================
extern "C" void kernel_launch(void* const* d_in, const int* in_sizes, int n_in,
                              void* d_out, int out_size, void* d_ws, size_t ws_size,
                              hipStream_t stream) {
    const float* x   = (const float*)d_in[0];   // [N, 64] float32
    const int*   src = (const int*)d_in[1];     // [E] int32
    const int*   tgt = (const int*)d_in[2];     // [E] int32
    float*       out = (float*)d_out;           // [N, 64] float32

    const int n_edges = in_sizes[1];
    const int n_nodes = out_size / F_DIM;
    const int n4      = out_size / 4;

    const int threads  = 256;
    const int n_chunks = (n_nodes + SCAN_CHUNK - 1) / SCAN_CHUNK;  // 98

    // ws layout (ints): cnt[N] | cursor[N] | rowptr[N] | chunk_sums[256] | csr[E]
    const size_t need = ((size_t)3 * n_nodes + 256 + (size_t)n_edges) * sizeof(int);

    if (ws_size >= need && n_chunks <= 256) {
        int* cnt    = (int*)d_ws;
        int* cursor = cnt + n_nodes;
        int* rowptr = cursor + n_nodes;
        int* csums  = rowptr + n_nodes;
        int* csr    = csums + 256;

        int grid_n = (n_nodes + threads - 1) / threads;
        int grid_e = (n_edges + threads - 1) / threads;

        // 1) zero counts + cursors
        mp_zero_kernel<<<grid_n, threads, 0, stream>>>(cnt, cursor, n_nodes);
        // 2) histogram of targets
        mp_count_kernel<<<grid_e, threads, 0, stream>>>(tgt, cnt, n_edges);
        // 3-5) exclusive scan -> rowptr
        mp_scan_chunk_kernel<<<n_chunks, SCAN_CHUNK, 0, stream>>>(
            cnt, rowptr, csums, n_nodes);
        mp_scan_sums_kernel<<<1, 256, 0, stream>>>(csums, n_chunks);
        mp_scan_add_kernel<<<n_chunks, SCAN_CHUNK, 0, stream>>>(
            rowptr, csums, n_nodes);
        // 6) bucket edges into CSR
        mp_fill_kernel<<<grid_e, threads, 0, stream>>>(
            src, tgt, rowptr, cursor, csr, n_edges);
        // 7) per-node gather + mean (one wave32 per node)
        long long gthreads = (long long)n_nodes * 32;
        int grid_g = (int)((gthreads + threads - 1) / threads);
        mp_gather_kernel<<<grid_g, threads, 0, stream>>>(
            x, csr, rowptr, cnt, out, n_nodes);
    } else {
        // Fallback: float-atomic scatter (needs only N ints of scratch).
        int* cnt = (int*)d_ws;
        int grid_init = (n4 + threads - 1) / threads;
        mp_init_kernel<<<grid_init, threads, 0, stream>>>(
            (float4*)out, cnt, n4, n_nodes);
        int grid_edges = (n_edges + threads - 1) / threads;
        mp_edge_scatter_kernel<<<grid_edges, threads, 0, stream>>>(
            x, src, tgt, out, cnt, n_edges);
        mp_finalize_kernel<<<grid_init, threads, 0, stream>>>(
            (float4*)out, cnt, n4);
    }
}